// GraphGenerator_38500086841660
// MI455X (gfx1250) — compile-verified
//
#include <hip/hip_runtime.h>
#include <hip/hip_bf16.h>
#include <math.h>

// ---------------- problem constants ----------------
#define N_GRAPH 100000
#define N_CAND  1000
#define NNODE   101000          // N_GRAPH + N_CAND
#define NEDGE   3200000
#define NPAD    101008          // ceil(NNODE/16)*16
#define RB      256             // reduction blocks
#define EPB     256             // edges per block in aggregation (NEDGE % EPB == 0)

typedef __attribute__((ext_vector_type(2))) float v2f;
typedef __attribute__((ext_vector_type(8))) float v8f;

// ---------------- async-to-LDS support (CDNA5 ASYNCcnt path) ----------------
#if defined(__has_builtin)
#  if __has_builtin(__builtin_amdgcn_global_load_async_to_lds_b32)
#    define HAVE_ASYNC_LDS 1
#  endif
#endif

typedef __attribute__((address_space(1))) int g1int;   // global addrspace
typedef __attribute__((address_space(3))) int l3int;   // LDS addrspace

#if defined(__has_builtin)
#  if __has_builtin(__builtin_amdgcn_s_wait_asynccnt)
#    define WAIT_ASYNC() __builtin_amdgcn_s_wait_asynccnt(0)
#  else
#    define WAIT_ASYNC() asm volatile("s_wait_asynccnt 0x0" ::: "memory")
#  endif
#else
#  define WAIT_ASYNC() asm volatile("s_wait_asynccnt 0x0" ::: "memory")
#endif

// ---------------- small helpers ----------------
__device__ __forceinline__ unsigned hash2(unsigned a, unsigned b) {
    unsigned x = a * 0x9E3779B9u ^ (b + 0x85EBCA6Bu);
    x ^= x >> 16; x *= 0x7feb352du;
    x ^= x >> 15; x *= 0x846ca68bu;
    x ^= x >> 16;
    return x;
}

// ---------------- build padded input h = concat(x, cand) ----------------
__global__ void kinit_h(const float* __restrict__ x, const float* __restrict__ cand,
                        float* __restrict__ H0) {
    int idx = blockIdx.x * blockDim.x + threadIdx.x;
    if (idx >= NPAD * 32) return;
    int i = idx >> 5, f = idx & 31;
    float v = 0.f;
    if (i < N_GRAPH)      v = x[(size_t)i * 32 + f];
    else if (i < NNODE)   v = cand[(size_t)(i - N_GRAPH) * 32 + f];
    H0[idx] = v;
}

// ---------------- degree / dinv ----------------
__global__ void kdeg_init(float* __restrict__ deg) {
    int i = blockIdx.x * blockDim.x + threadIdx.x;
    if (i < NPAD) deg[i] = 1.0f;   // self-loop; pads get 1 so rsqrt stays finite
}
__global__ void kdeg_count(const int* __restrict__ colp, float* __restrict__ deg) {
    int e = blockIdx.x * blockDim.x + threadIdx.x;
    if (e < NEDGE) {
        __builtin_prefetch(&colp[e + 8192], 0, 0);   // stream edge list ahead
        atomicAdd(&deg[colp[e]], 1.0f);
    }
}
__global__ void kdinv(float* __restrict__ deg) {
    int i = blockIdx.x * blockDim.x + threadIdx.x;
    if (i < NPAD) deg[i] = rsqrtf(deg[i]);
}

// ---------------- dense transform hW = H[:, :K] @ W  via f32 WMMA ----------------
// One wave computes a 16-row tile across all output columns.
// A frag (16x4 f32, 2 VGPRs): lanes 0-15 hold K=kk+0/kk+1, lanes 16-31 hold K=kk+2/kk+3.
// B frag (4x16 f32, 2 VGPRs): mirrored row striping.
// C/D 16x16 f32 in 8 VGPRs: lanes 0-15 -> M=m0+reg, N=lane; lanes 16-31 -> M=m0+8+reg.
template<int K, int NOUT>
__global__ __launch_bounds__(128)
void kgemm_wmma(const float* __restrict__ H, const float* __restrict__ W,
                float* __restrict__ HW) {
    const int lane = threadIdx.x & 31;
    const int wv   = threadIdx.x >> 5;
    const int tile = blockIdx.x * 4 + wv;
    const int m0   = tile * 16;
    if (m0 >= NPAD) return;

    const int half = lane >> 4;      // 0: lanes 0-15, 1: lanes 16-31
    const int l    = lane & 15;
    constexpr int CT = (NOUT + 15) / 16;

    v8f acc[CT];
    #pragma unroll
    for (int c = 0; c < CT; ++c) {
        v8f z = {0.f, 0.f, 0.f, 0.f, 0.f, 0.f, 0.f, 0.f};
        acc[c] = z;
    }

    const float* arow = H + (size_t)(m0 + l) * 32;

    #pragma unroll
    for (int kk = 0; kk < K; kk += 4) {
        v2f a;
        a.x = arow[kk + 2 * half + 0];
        a.y = arow[kk + 2 * half + 1];
        #pragma unroll
        for (int c = 0; c < CT; ++c) {
            int col = c * 16 + l;
            float b0 = 0.f, b1 = 0.f;
            if (col < NOUT) {
                b0 = W[(kk + 2 * half + 0) * NOUT + col];
                b1 = W[(kk + 2 * half + 1) * NOUT + col];
            }
            v2f b; b.x = b0; b.y = b1;
            acc[c] = __builtin_amdgcn_wmma_f32_16x16x4_f32(
                /*neg_a=*/false, a, /*neg_b=*/false, b,
                /*c_mod=*/(short)0, acc[c], /*reuse_a=*/false, /*reuse_b=*/false);
        }
    }

    #pragma unroll
    for (int c = 0; c < CT; ++c) {
        int col = c * 16 + l;
        if (col < NOUT) {
            float* o = HW + (size_t)(m0 + half * 8) * 32 + col;
            #pragma unroll
            for (int i = 0; i < 8; ++i) o[(size_t)i * 32] = acc[c][i];
        }
    }
}

// ---------------- aggregation: OUT = D^-1/2 (A+I) D^-1/2 (hW) + b ----------------
template<int F>
__global__ void kagg_init(float* __restrict__ OUT, const float* __restrict__ HW,
                          const float* __restrict__ dinv, const float* __restrict__ bias) {
    int idx = blockIdx.x * blockDim.x + threadIdx.x;
    if (idx >= NPAD * 32) return;
    int i = idx >> 5, f = idx & 31;
    float v = 0.f;
    if (i < NNODE && f < F) {
        float di = dinv[i];
        v = HW[idx] * di * di + bias[f];   // self-loop + bias
    }
    OUT[idx] = v;
}

// Each block stages EPB=256 edges of the index stream into LDS (async when the
// toolchain exposes GLOBAL_LOAD_ASYNC_TO_LDS), then processes all EPB*F
// (edge,feature) pairs. Pair ordering p = j*EPB + tid keeps a full wave on one
// edge row for F=32 -> one cacheline per gather / per atomic burst, and the
// feature table (13 MB) is L2-resident on MI455X's 192 MB L2.
template<int F>
__global__ __launch_bounds__(EPB)
void kagg_edges(float* __restrict__ OUT, const float* __restrict__ HW,
                const float* __restrict__ dinv,
                const int* __restrict__ rowp, const int* __restrict__ colp) {
    __shared__ int srow[EPB];
    __shared__ int scol[EPB];
    const int tid = threadIdx.x;
    const int e0  = blockIdx.x * EPB;

#if defined(HAVE_ASYNC_LDS)
    __builtin_amdgcn_global_load_async_to_lds_b32(
        (g1int*)(rowp + e0 + tid), (l3int*)&srow[tid], 0, 0);
    __builtin_amdgcn_global_load_async_to_lds_b32(
        (g1int*)(colp + e0 + tid), (l3int*)&scol[tid], 0, 0);
    WAIT_ASYNC();
#else
    srow[tid] = rowp[e0 + tid];
    scol[tid] = colp[e0 + tid];
#endif
    __syncthreads();

    #pragma unroll 4
    for (int j = 0; j < F; ++j) {
        int p  = j * EPB + tid;
        int el = p / F;
        int f  = p - el * F;
        int r  = srow[el];
        int c  = scol[el];
        float v = HW[(size_t)r * 32 + f] * dinv[r] * dinv[c];
        atomicAdd(&OUT[(size_t)c * 32 + f], v);
    }
}

// ---------------- MLP heads: logits = relu6(h@W1+b1)@W2+b2 ----------------
template<int HID>
__global__ void khead(const float* __restrict__ H, const float* __restrict__ W1,
                      const float* __restrict__ B1, const float* __restrict__ W2,
                      const float* __restrict__ B2, float* __restrict__ logits) {
    int i = blockIdx.x * blockDim.x + threadIdx.x;
    if (i >= NNODE) return;
    const float* hr = H + (size_t)i * 32;
    float h[32];
    #pragma unroll
    for (int k = 0; k < 32; ++k) h[k] = hr[k];
    float acc = B2[0];
    #pragma unroll
    for (int j = 0; j < HID; ++j) {
        float s = B1[j];
        #pragma unroll
        for (int k = 0; k < 32; ++k) s += h[k] * W1[k * HID + j];
        s = fminf(fmaxf(s, 0.f), 6.f);
        acc += s * W2[j];
    }
    logits[i] = acc;
}

// ---------------- softmax reductions ----------------
__global__ void kmax_part(const float* __restrict__ v, float* __restrict__ part) {
    __shared__ float s[256];
    float m = -INFINITY;
    for (int i = blockIdx.x * 256 + threadIdx.x; i < NNODE; i += RB * 256)
        m = fmaxf(m, v[i]);
    s[threadIdx.x] = m; __syncthreads();
    for (int o = 128; o > 0; o >>= 1) {
        if (threadIdx.x < o) s[threadIdx.x] = fmaxf(s[threadIdx.x], s[threadIdx.x + o]);
        __syncthreads();
    }
    if (threadIdx.x == 0) part[blockIdx.x] = s[0];
}
__global__ void kmax_final(const float* __restrict__ part, float* __restrict__ sc) {
    __shared__ float s[256];
    s[threadIdx.x] = part[threadIdx.x]; __syncthreads();
    for (int o = 128; o > 0; o >>= 1) {
        if (threadIdx.x < o) s[threadIdx.x] = fmaxf(s[threadIdx.x], s[threadIdx.x + o]);
        __syncthreads();
    }
    if (threadIdx.x == 0) sc[0] = s[0];
}
__global__ void kexp_part(const float* __restrict__ v, const float* __restrict__ sc,
                          float* __restrict__ expu, float* __restrict__ part) {
    __shared__ float s[256];
    float mx = sc[0];
    float acc = 0.f;
    for (int i = blockIdx.x * 256 + threadIdx.x; i < NNODE; i += RB * 256) {
        float e = expf(v[i] - mx);
        expu[i] = e;
        acc += e;
    }
    s[threadIdx.x] = acc; __syncthreads();
    for (int o = 128; o > 0; o >>= 1) {
        if (threadIdx.x < o) s[threadIdx.x] += s[threadIdx.x + o];
        __syncthreads();
    }
    if (threadIdx.x == 0) part[blockIdx.x] = s[0];
}
__global__ void ksum_final(const float* __restrict__ part, float* __restrict__ sc) {
    __shared__ float s[256];
    s[threadIdx.x] = part[threadIdx.x]; __syncthreads();
    for (int o = 128; o > 0; o >>= 1) {
        if (threadIdx.x < o) s[threadIdx.x] += s[threadIdx.x + o];
        __syncthreads();
    }
    if (threadIdx.x == 0) sc[1] = s[0];
}
__global__ void kfin_start(const float* __restrict__ expu, const float* __restrict__ sc,
                           float* __restrict__ probs) {
    int i = blockIdx.x * blockDim.x + threadIdx.x;
    if (i >= NNODE) return;
    float p = expu[i] / sc[1];
    if (i >= N_GRAPH) p = 0.f;          // mask candidate rows
    if (p == 0.f) p = 1e-10f;
    probs[i] = p;
}
__global__ void kfin_end(const float* __restrict__ expu, const float* __restrict__ sc,
                         const int* __restrict__ nodes, float* __restrict__ probs) {
    int i = blockIdx.x * blockDim.x + threadIdx.x;
    if (i >= NNODE) return;
    float p = expu[i] / sc[1];
    if (i == nodes[0]) p = 0.f;         // zero out start node
    if (p == 0.f) p = 1e-10f;
    probs[i] = p;
}

// ---------------- deterministic Gumbel-argmax "categorical" ----------------
__global__ void kargmax_part(const float* __restrict__ p, unsigned seed,
                             float* __restrict__ pv, int* __restrict__ pi) {
    __shared__ float sv[256]; __shared__ int si[256];
    float best = -INFINITY; int bi = 0;
    for (int i = blockIdx.x * 256 + threadIdx.x; i < NNODE; i += RB * 256) {
        unsigned hx = hash2(seed, (unsigned)i);
        float u = ((float)hx + 0.5f) * (1.0f / 4294967296.0f);
        float g = -logf(-logf(u));
        float s = logf(p[i]) + g;
        if (s > best) { best = s; bi = i; }
    }
    sv[threadIdx.x] = best; si[threadIdx.x] = bi; __syncthreads();
    for (int o = 128; o > 0; o >>= 1) {
        if (threadIdx.x < o && sv[threadIdx.x + o] > sv[threadIdx.x]) {
            sv[threadIdx.x] = sv[threadIdx.x + o];
            si[threadIdx.x] = si[threadIdx.x + o];
        }
        __syncthreads();
    }
    if (threadIdx.x == 0) { pv[blockIdx.x] = sv[0]; pi[blockIdx.x] = si[0]; }
}
__global__ void kargmax_final(const float* __restrict__ pv, const int* __restrict__ pi,
                              int* __restrict__ nodes, int which,
                              float* __restrict__ outscalar) {
    __shared__ float sv[256]; __shared__ int si[256];
    sv[threadIdx.x] = pv[threadIdx.x]; si[threadIdx.x] = pi[threadIdx.x]; __syncthreads();
    for (int o = 128; o > 0; o >>= 1) {
        if (threadIdx.x < o && sv[threadIdx.x + o] > sv[threadIdx.x]) {
            sv[threadIdx.x] = sv[threadIdx.x + o];
            si[threadIdx.x] = si[threadIdx.x + o];
        }
        __syncthreads();
    }
    if (threadIdx.x == 0) { nodes[which] = si[0]; *outscalar = (float)si[0]; }
}

// ---------------- output copies ----------------
__global__ void kcopy_h(const float* __restrict__ H, float* __restrict__ out) {
    int i = blockIdx.x * blockDim.x + threadIdx.x;
    if (i < NNODE * 32) out[i] = H[i];
}
__global__ void kcopy_edges(const int* __restrict__ edges, float* __restrict__ out) {
    int i = blockIdx.x * blockDim.x + threadIdx.x;
    if (i >= 2 * NEDGE) return;
    int r = (i < NEDGE) ? 0 : 1;
    int pos = i - r * NEDGE;
    out[(size_t)r * (NEDGE + 1) + pos] = (float)edges[i];
}
__global__ void kfix_edges(const int* __restrict__ nodes, float* __restrict__ out) {
    if (threadIdx.x == 0 && blockIdx.x == 0) {
        out[NEDGE] = (float)nodes[0];
        out[(size_t)2 * NEDGE + 1] = (float)nodes[1];
    }
}

// ---------------- launcher ----------------
extern "C" void kernel_launch(void* const* d_in, const int* in_sizes, int n_in,
                              void* d_out, int out_size, void* d_ws, size_t ws_size,
                              hipStream_t stream) {
    const float* x    = (const float*)d_in[0];
    const float* cand = (const float*)d_in[1];
    const int*   edge = (const int*)  d_in[2];
    const float* W1 = (const float*)d_in[3];  const float* b1 = (const float*)d_in[4];
    const float* W2 = (const float*)d_in[5];  const float* b2 = (const float*)d_in[6];
    const float* W3 = (const float*)d_in[7];  const float* b3 = (const float*)d_in[8];
    const float* Ws1 = (const float*)d_in[9];  const float* bs1 = (const float*)d_in[10];
    const float* Ws2 = (const float*)d_in[11]; const float* bs2 = (const float*)d_in[12];
    const float* We1 = (const float*)d_in[13]; const float* be1 = (const float*)d_in[14];
    const float* We2 = (const float*)d_in[15]; const float* be2 = (const float*)d_in[16];

    const int* rowp = edge;
    const int* colp = edge + NEDGE;

    // workspace layout (floats)
    float* ws   = (float*)d_ws;
    float* H0   = ws;
    float* HW   = H0 + (size_t)NPAD * 32;
    float* HN   = HW + (size_t)NPAD * 32;
    float* DINV = HN + (size_t)NPAD * 32;
    float* LOGI = DINV + NPAD;
    float* EXPU = LOGI + NNODE;
    float* PV   = EXPU + NNODE;
    int*   PI   = (int*)(PV + RB);
    float* SC   = (float*)(PI + RB);     // [0]=max, [1]=sum
    int*   NODES = (int*)(SC + 8);

    float* dout = (float*)d_out;
    float* O_H  = dout;                              // [NNODE*32]
    float* O_SP = dout + (size_t)NNODE * 32;         // start_probs [NNODE]
    float* O_EP = O_SP + NNODE;                      // end_probs   [NNODE]
    float* O_SN = O_EP + NNODE;                      // start_node  [1]
    float* O_EN = O_SN + 1;                          // end_node    [1]
    float* O_E  = O_EN + 1;                          // edge_index_new [2, NEDGE+1]

    const int T = 256;
    const int gHF   = (NPAD * 32 + T - 1) / T;
    const int gN    = (NPAD + T - 1) / T;
    const int gNode = (NNODE + T - 1) / T;
    const int gE    = (NEDGE + T - 1) / T;
    const int gTile = ((NPAD / 16) + 3) / 4;
    const int gAgg  = NEDGE / EPB;                   // 12500, exact

    // 1) concat inputs into padded stride-32 buffer
    kinit_h<<<gHF, T, 0, stream>>>(x, cand, H0);

    // 2) degrees -> dinv
    kdeg_init<<<gN, T, 0, stream>>>(DINV);
    kdeg_count<<<gE, T, 0, stream>>>(colp, DINV);
    kdinv<<<gN, T, 0, stream>>>(DINV);

    // 3) GCN layer 1: 32 -> 16   (H0 -> HW -> HN)
    kgemm_wmma<32, 16><<<gTile, 128, 0, stream>>>(H0, W1, HW);
    kagg_init<16><<<gHF, T, 0, stream>>>(HN, HW, DINV, b1);
    kagg_edges<16><<<gAgg, EPB, 0, stream>>>(HN, HW, DINV, rowp, colp);

    // 4) GCN layer 2: 16 -> 24   (HN -> HW -> H0)
    kgemm_wmma<16, 24><<<gTile, 128, 0, stream>>>(HN, W2, HW);
    kagg_init<24><<<gHF, T, 0, stream>>>(H0, HW, DINV, b2);
    kagg_edges<24><<<gAgg, EPB, 0, stream>>>(H0, HW, DINV, rowp, colp);

    // 5) GCN layer 3: 24 -> 32   (H0 -> HW -> HN)
    kgemm_wmma<24, 32><<<gTile, 128, 0, stream>>>(H0, W3, HW);
    kagg_init<32><<<gHF, T, 0, stream>>>(HN, HW, DINV, b3);
    kagg_edges<32><<<gAgg, EPB, 0, stream>>>(HN, HW, DINV, rowp, colp);

    // 6) output h
    kcopy_h<<<(NNODE * 32 + T - 1) / T, T, 0, stream>>>(HN, O_H);

    // 7) start head: logits -> softmax -> mask -> sample
    khead<16><<<gNode, T, 0, stream>>>(HN, Ws1, bs1, Ws2, bs2, LOGI);
    kmax_part<<<RB, 256, 0, stream>>>(LOGI, PV);
    kmax_final<<<1, 256, 0, stream>>>(PV, SC);
    kexp_part<<<RB, 256, 0, stream>>>(LOGI, SC, EXPU, PV);
    ksum_final<<<1, 256, 0, stream>>>(PV, SC);
    kfin_start<<<gNode, T, 0, stream>>>(EXPU, SC, O_SP);
    kargmax_part<<<RB, 256, 0, stream>>>(O_SP, 42u, PV, PI);
    kargmax_final<<<1, 256, 0, stream>>>(PV, PI, NODES, 0, O_SN);

    // 8) end head: logits -> softmax -> zero(start) -> sample
    khead<24><<<gNode, T, 0, stream>>>(HN, We1, be1, We2, be2, LOGI);
    kmax_part<<<RB, 256, 0, stream>>>(LOGI, PV);
    kmax_final<<<1, 256, 0, stream>>>(PV, SC);
    kexp_part<<<RB, 256, 0, stream>>>(LOGI, SC, EXPU, PV);
    ksum_final<<<1, 256, 0, stream>>>(PV, SC);
    kfin_end<<<gNode, T, 0, stream>>>(EXPU, SC, NODES, O_EP);
    kargmax_part<<<RB, 256, 0, stream>>>(O_EP, 43u, PV, PI);
    kargmax_final<<<1, 256, 0, stream>>>(PV, PI, NODES, 1, O_EN);

    // 9) edge_index_new = [edge_index | (start,end)]
    kcopy_edges<<<(2 * NEDGE + T - 1) / T, T, 0, stream>>>(edge, O_E);
    kfix_edges<<<1, 64, 0, stream>>>(NODES, O_E);
}